// FeatureInterpolator_47571057771107
// MI455X (gfx1250) — compile-verified
//
#include <hip/hip_runtime.h>

#define BB 4
#define N1 16384
#define N2 4096
#define C1 128
#define C2 256
#define CO (C1 + C2)   // 384
#define EPSV 1e-10f

typedef __attribute__((ext_vector_type(2))) float v2f;
typedef __attribute__((ext_vector_type(4))) float v4f;
typedef __attribute__((ext_vector_type(8))) float v8f;

__device__ __forceinline__ bool any32(bool c) {
    return __builtin_amdgcn_ballot_w32(c) != 0u;
}

// Raw min/max: avoids clang's sNaN-canonicalize (v_max_num x,x,x) around fminf.
// Safe here: all inputs are finite or +INF, never NaN.
__device__ __forceinline__ float fmin_raw(float a, float b) {
    float d;
    asm("v_min_num_f32 %0, %1, %2" : "=v"(d) : "v"(a), "v"(b));
    return d;
}
__device__ __forceinline__ float fmax_raw(float a, float b) {
    float d;
    asm("v_max_num_f32 %0, %1, %2" : "=v"(d) : "v"(a), "v"(b));
    return d;
}

__device__ __forceinline__ v8f wmma4(v2f A, v2f Bv) {
    v8f cz = {};
    return __builtin_amdgcn_wmma_f32_16x16x4_f32(false, A, false, Bv, (short)0, cz,
                                                 false, false);
}

// Value-only sorted-3 insert: keep 3 smallest (min/max network, 5 VALU).
__device__ __forceinline__ void vins(float v, float& s0, float& s1, float& s2) {
    float n0 = fmin_raw(s0, v), h = fmax_raw(s0, v);
    float n1 = fmin_raw(s1, h), h2 = fmax_raw(s1, h);
    float n2 = fmin_raw(s2, h2);
    s0 = n0; s1 = n1; s2 = n2;
}

// Exact (value,index) sorted-3 insert; strict '<' => earlier index wins ties.
__device__ __forceinline__ void insert3(float v, int m, float (&s)[3], int (&ix)[3]) {
    bool c2 = v < s[2], c1 = v < s[1], c0 = v < s[0];
    float t2 = c1 ? s[1] : (c2 ? v : s[2]);  int j2 = c1 ? ix[1] : (c2 ? m : ix[2]);
    float t1 = c0 ? s[0] : (c1 ? v : s[1]);  int j1 = c0 ? ix[0] : (c1 ? m : ix[1]);
    float t0 = c0 ? v    : s[0];             int j0 = c0 ? m    : ix[0];
    s[0] = t0; s[1] = t1; s[2] = t2;
    ix[0] = j0; ix[1] = j1; ix[2] = j2;
}

// ---- Kernel 0: ym2[b][m] = ||sparse_xyz[b,:,m]||^2 ----
__global__ void k_ym2(const float* __restrict__ sx, float* __restrict__ ym2) {
    int gid = blockIdx.x * blockDim.x + threadIdx.x;   // B*N2 threads
    int b = gid >> 12;
    int m = gid & (N2 - 1);
    const float* p = sx + (size_t)b * 3 * N2 + m;
    float y0 = p[0], y1 = p[N2], y2 = p[2 * N2];
    ym2[gid] = y0 * y0 + y1 * y1 + y2 * y2;
}

// ---- Kernel 1: exact 3-NN via WMMA, two passes. One wave = 16 dense rows. ----
__global__ void __launch_bounds__(256)
k_knn(const float* __restrict__ dx, const float* __restrict__ sx,
      const float* __restrict__ ym2, int* __restrict__ outIdx) {
    const int tid  = threadIdx.x;
    const int lane = tid & 31;
    const int wid  = tid >> 5;
    const int rt   = blockIdx.x * 8 + wid;      // 0..4095 row-tiles
    const int b    = rt >> 10;                  // / (N1/16)
    const int n0   = (rt & 1023) << 4;
    const bool lo  = lane < 16;
    const int c16  = lane & 15;
    const int NT   = N2 / 16;                   // 256 column tiles
    const float INF = __builtin_inff();

    // A (16x4 f32): rows = dense points, cols (-2x0,-2x1,-2x2, 1).
    // VGPR0: K=0 (lanes 0-15) / K=2 (lanes 16-31); VGPR1: K=1 / K=3.
    const float* dxb = dx + (size_t)b * 3 * N1;
    float a0 = -2.0f * dxb[(lo ? 0 : 2) * N1 + n0 + c16];
    float a1r = dxb[1 * N1 + n0 + c16];
    v2f A; A[0] = a0; A[1] = lo ? (-2.0f * a1r) : 1.0f;

    // B (4x16 f32) per column tile: rows (y0, y1, y2, ||y||^2).
    const float* sxb = sx + (size_t)b * 3 * N2;
    const float* base0 = sxb + (lo ? 0 : 2) * N2 + c16;
    const float* base1 = lo ? (sxb + N2 + c16) : (ym2 + (size_t)b * N2 + c16);

    // ---------------- Pass 1: per-lane running min per row ----------------
    float mn[8];
#pragma unroll
    for (int r = 0; r < 8; ++r) mn[r] = INF;
    {
        const float* p0 = base0;
        const float* p1 = base1;
        v2f cur; cur[0] = *p0; cur[1] = *p1;
        for (int mt = 0; mt < NT; ++mt) {
            v2f nxt = cur;
            if (mt + 1 < NT) { nxt[0] = p0[16]; nxt[1] = p1[16]; }  // prefetch
            p0 += 16; p1 += 16;
            v8f acc = wmma4(A, cur);    // key = ||y||^2 - 2 x.y (rank == d2 rank)
#pragma unroll
            for (int r = 0; r < 8; ++r) mn[r] = fmin_raw(mn[r], acc[r]);
            cur = nxt;
        }
    }
    // Threshold T[r] = 3rd smallest of the 16 per-lane minima (distinct columns),
    // hence T[r] >= true 3rd-smallest key of the row. Value-only butterfly.
    float T[8];
    {
        float t0[8], t1[8], t2[8];
#pragma unroll
        for (int r = 0; r < 8; ++r) { t0[r] = mn[r]; t1[r] = INF; t2[r] = INF; }
#pragma unroll
        for (int step = 1; step <= 8; step <<= 1) {
#pragma unroll
            for (int r = 0; r < 8; ++r) {
                float o0 = __shfl_xor(t0[r], step, 32);
                float o1 = __shfl_xor(t1[r], step, 32);
                float o2 = __shfl_xor(t2[r], step, 32);
                vins(o0, t0[r], t1[r], t2[r]);
                vins(o1, t0[r], t1[r], t2[r]);
                vins(o2, t0[r], t1[r], t2[r]);
            }
        }
#pragma unroll
        for (int r = 0; r < 8; ++r) T[r] = t2[r];
    }

    // ---------------- Pass 2: collect indices of keys <= T ----------------
    float sv[8][3];
    int   si[8][3];
#pragma unroll
    for (int r = 0; r < 8; ++r) {
        sv[r][0] = sv[r][1] = sv[r][2] = INF;
        si[r][0] = si[r][1] = si[r][2] = 0;
    }
    {
        const float* p0 = base0;
        const float* p1 = base1;
        v2f cur; cur[0] = *p0; cur[1] = *p1;
        int mcur = c16;
        for (int mt = 0; mt < NT; ++mt) {
            v2f nxt = cur;
            if (mt + 1 < NT) { nxt[0] = p0[16]; nxt[1] = p1[16]; }
            p0 += 16; p1 += 16;
            v8f acc = wmma4(A, cur);    // bit-identical keys to pass 1
#pragma unroll
            for (int r = 0; r < 8; ++r) {
                float v = acc[r];
                if (any32(v <= T[r]))               // uniform branch, rarely taken
                    insert3(v, mcur, sv[r], si[r]);
            }
            mcur += 16;
            cur = nxt;
        }
    }
    // Exact (val,idx) butterfly merge within each 16-lane half.
#pragma unroll
    for (int step = 1; step <= 8; step <<= 1) {
#pragma unroll
        for (int r = 0; r < 8; ++r) {
            float o0 = __shfl_xor(sv[r][0], step, 32);
            float o1 = __shfl_xor(sv[r][1], step, 32);
            float o2 = __shfl_xor(sv[r][2], step, 32);
            int   j0 = __shfl_xor(si[r][0], step, 32);
            int   j1 = __shfl_xor(si[r][1], step, 32);
            int   j2 = __shfl_xor(si[r][2], step, 32);
            insert3(o0, j0, sv[r], si[r]);
            insert3(o1, j1, sv[r], si[r]);
            insert3(o2, j2, sv[r], si[r]);
        }
    }

    if ((lane & 15) == 0) {
        int half = lane >> 4;                   // 0 -> rows 0-7, 1 -> rows 8-15
#pragma unroll
        for (int r = 0; r < 8; ++r) {
            int n = n0 + half * 8 + r;
            int* q = outIdx + ((size_t)b * N1 + n) * 4;
            q[0] = si[r][0]; q[1] = si[r][1]; q[2] = si[r][2]; q[3] = 0;
        }
    }
}

// ---- Kernel 2: exact distances -> normalized inverse-distance weights ----
__global__ void k_weights(const float* __restrict__ dx, const float* __restrict__ sx,
                          const int* __restrict__ idx4, float* __restrict__ w4) {
    int gid = blockIdx.x * blockDim.x + threadIdx.x;   // B*N1 threads
    int b = gid >> 14;
    int n = gid & (N1 - 1);
    const float* dxb = dx + (size_t)b * 3 * N1 + n;
    float x0 = dxb[0], x1 = dxb[N1], x2 = dxb[2 * N1];
    const float* sxb = sx + (size_t)b * 3 * N2;
    const int* q = idx4 + (size_t)gid * 4;
    float inv[3];
#pragma unroll
    for (int k = 0; k < 3; ++k) {
        int m = q[k];
        float d0 = x0 - sxb[m];
        float d1 = x1 - sxb[N2 + m];
        float d2c = x2 - sxb[2 * N2 + m];
        float d2 = d0 * d0 + d1 * d1 + d2c * d2c;
        d2 = fmax_raw(d2, EPSV);
        inv[k] = 1.0f / d2;
    }
    float s = inv[0] + inv[1] + inv[2];
    float* w = w4 + (size_t)gid * 4;
    w[0] = inv[0] / s; w[1] = inv[1] / s; w[2] = inv[2] / s; w[3] = 0.0f;
}

// ---- Kernel 3: weighted gather of sparse features (channels 0..255) ----
__global__ void __launch_bounds__(256)
k_interp(const float* __restrict__ sf, const int* __restrict__ idx4,
         const float* __restrict__ w4, float* __restrict__ out) {
    int n  = blockIdx.x * 256 + threadIdx.x;
    int c0 = blockIdx.y * 8;
    int b  = blockIdx.z;
    size_t bn = (size_t)b * N1 + n;
    const int4   iv = ((const int4*)idx4)[bn];
    const float4 wv = ((const float4*)w4)[bn];
    const float* sfb = sf + (size_t)b * C2 * N2;
    float* ob = out + (size_t)b * CO * N1 + (size_t)c0 * N1 + n;
#pragma unroll
    for (int cc = 0; cc < 8; ++cc) {
        const float* row = sfb + (size_t)(c0 + cc) * N2;
        float v = wv.x * row[iv.x] + wv.y * row[iv.y] + wv.z * row[iv.z];
        __builtin_nontemporal_store(v, ob + (size_t)cc * N1);   // streaming output
    }
}

// ---- Kernel 4: copy dense features into channels 256..383 (float4, NT) ----
__global__ void k_copy(const float* __restrict__ df, float* __restrict__ out) {
    int f = blockIdx.x * 256 + threadIdx.x;     // B*C1*N1/4 float4 elems
    int b = f / (C1 * N1 / 4);
    int r = f - b * (C1 * N1 / 4);
    int c = r >> 12;                            // / (N1/4)
    int n4 = r & 4095;
    v4f val = __builtin_nontemporal_load((const v4f*)df + f);
    __builtin_nontemporal_store(val, (v4f*)out + ((size_t)(b * CO + C2 + c) << 12) + n4);
}

extern "C" void kernel_launch(void* const* d_in, const int* in_sizes, int n_in,
                              void* d_out, int out_size, void* d_ws, size_t ws_size,
                              hipStream_t stream) {
    const float* dx = (const float*)d_in[0];   // dense_xyz  (B,3,N1)
    const float* sx = (const float*)d_in[1];   // sparse_xyz (B,3,N2)
    const float* df = (const float*)d_in[2];   // dense_feature (B,C1,N1)
    const float* sf = (const float*)d_in[3];   // sparse_feature (B,C2,N2)
    float* out = (float*)d_out;                // (B, C2+C1, N1)

    // ws layout: ym2 (B*N2 f32 = 64KB) | idx4 (B*N1*4 i32 = 1MB) | w4 (B*N1*4 f32 = 1MB)
    float* ym2  = (float*)d_ws;
    int*   idx4 = (int*)((char*)d_ws + (size_t)BB * N2 * sizeof(float));
    float* w4   = (float*)((char*)idx4 + (size_t)BB * N1 * 4 * sizeof(int));

    k_ym2    <<<BB * N2 / 256, 256, 0, stream>>>(sx, ym2);
    k_knn    <<<BB * (N1 / 16) / 8, 256, 0, stream>>>(dx, sx, ym2, idx4);
    k_weights<<<BB * N1 / 256, 256, 0, stream>>>(dx, sx, idx4, w4);
    k_interp <<<dim3(N1 / 256, C2 / 8, BB), 256, 0, stream>>>(sf, idx4, w4, out);
    k_copy   <<<BB * C1 * N1 / 4 / 256, 256, 0, stream>>>(df, out);
}